// EdgeConvE_74268574482771
// MI455X (gfx1250) — compile-verified
//
#include <hip/hip_runtime.h>

typedef float v2f __attribute__((ext_vector_type(2)));
typedef float v8f __attribute__((ext_vector_type(8)));

#define N_NODES 1024
#define F_NODE  32
#define E_ATTR  8
#define H_OUT   64
#define NTILE   16                                   // neighbors per tile
#define HTILES  4                                    // 4 x 16 = H_OUT
#define WAVES   4
#define TILES_PER_WAVE (N_NODES / NTILE / WAVES)     // 16

// Kernel 1: self_term[n,h] = X[n,:] @ (W_self - W_diff)[:,h] + b[h]
//           nbr_term[n,h]  = X[n,:] @ W_diff[:,h]
__global__ __launch_bounds__(256) void precompute_terms(
    const float* __restrict__ X, const float* __restrict__ W,
    const float* __restrict__ b, float* __restrict__ selfT,
    float* __restrict__ nbrT) {
  int idx = blockIdx.x * blockDim.x + threadIdx.x;
  if (idx >= N_NODES * H_OUT) return;
  int n = idx >> 6;   // / H_OUT
  int h = idx & 63;   // % H_OUT
  float sd = 0.f, nd = 0.f;
#pragma unroll
  for (int f = 0; f < F_NODE; ++f) {
    float x  = X[n * F_NODE + f];
    float wd = W[(F_NODE + f) * H_OUT + h];   // W_diff
    sd = fmaf(x, W[f * H_OUT + h] - wd, sd);  // W_self - W_diff
    nd = fmaf(x, wd, nd);
  }
  selfT[idx] = sd + b[h];   // fold bias into self term
  nbrT[idx]  = nd;
}

// Kernel 2: block per node v; wave q owns w-tiles in [q*256, (q+1)*256) and
// covers ALL 4 h-tiles per w-tile so the edge A-tile is loaded once and
// reused in registers across 4 WMMA pairs. Cross-wave partials combined in LDS.
__global__ __launch_bounds__(128) void edgeconv_wmma(
    const int* __restrict__ adj, const float* __restrict__ edge,
    const float* __restrict__ W, const float* __restrict__ selfT,
    const float* __restrict__ nbrT, float* __restrict__ out) {
  __shared__ float s_adj[N_NODES];
  __shared__ float s_part[WAVES][H_OUT];

  const int v    = blockIdx.x;
  const int tid  = threadIdx.x;
  const int wave = tid >> 5;
  const int lane = tid & 31;
  const int half = lane >> 4;   // lane half: K sub-pair for A/B, row group for C/D
  const int col  = lane & 15;   // A row M; B/C/D column N

  // Stage adjacency row for v into LDS as float weights.
  for (int i = tid; i < N_NODES; i += 128)
    s_adj[i] = (float)adj[v * N_NODES + i];
  __syncthreads();

  // Loop-invariant B operands (W_edge slices) + self term, per h-tile.
  const float* We = W + 2 * F_NODE * H_OUT;  // rows 64..71 of W
  const int k0 = 2 * half;
  v2f b0[HTILES], b1[HTILES];
  float sv[HTILES], acc[HTILES];
#pragma unroll
  for (int ht = 0; ht < HTILES; ++ht) {
    const int h = ht * 16 + col;
    b0[ht] = { We[(k0 + 0) * H_OUT + h], We[(k0 + 1) * H_OUT + h] };  // K 0..3
    b1[ht] = { We[(k0 + 4) * H_OUT + h], We[(k0 + 5) * H_OUT + h] };  // K 4..7
    sv[ht] = selfT[v * H_OUT + h];
    acc[ht] = 0.f;
  }

  const float* erow = edge + (size_t)v * N_NODES * E_ATTR;
  const int wb0 = wave * (N_NODES / WAVES);   // this wave's 256-neighbor span

  for (int t = 0; t < TILES_PER_WAVE; ++t) {
    const int wb = wb0 + t * NTILE;

    // A operand: 16x4 f32 edge tile, row M = col, lane half picks K pair.
    // Loaded ONCE per w-tile, reused by all 4 h-tiles.
    const float* ep = erow + (size_t)(wb + col) * E_ATTR + k0;
    v2f a0 = *(const v2f*)(ep);       // K 0..3 sub-block
    v2f a1 = *(const v2f*)(ep + 4);   // K 4..7 sub-block
    __builtin_prefetch(ep + NTILE * E_ATTR, 0, 1);  // next tile (speculative)

    // Adjacency weights for rows M = r + 8*half, shared by all h-tiles.
    float aw[8];
#pragma unroll
    for (int r = 0; r < 8; ++r)
      aw[r] = s_adj[wb + r + 8 * half];

#pragma unroll
    for (int ht = 0; ht < HTILES; ++ht) {
      const int h = ht * 16 + col;
      v8f c;
#pragma unroll
      for (int r = 0; r < 8; ++r)
        c[r] = sv[ht] + nbrT[(wb + r + 8 * half) * H_OUT + h];

      c = __builtin_amdgcn_wmma_f32_16x16x4_f32(false, a0, false, b0[ht],
                                                (short)0, c, false, false);
      c = __builtin_amdgcn_wmma_f32_16x16x4_f32(false, a1, false, b1[ht],
                                                (short)0, c, false, false);

#pragma unroll
      for (int r = 0; r < 8; ++r)
        acc[ht] = fmaf(fmaxf(c[r], 0.f), aw[r], acc[ht]);
    }
  }

  // Combine lane halves (L and L^16 hold the same h column), then waves.
#pragma unroll
  for (int ht = 0; ht < HTILES; ++ht) {
    float total = acc[ht] + __shfl_xor(acc[ht], 16, 32);
    if (lane < 16) s_part[wave][ht * 16 + col] = total;
  }
  __syncthreads();

  if (tid < H_OUT)
    out[v * H_OUT + tid] = (s_part[0][tid] + s_part[1][tid]) +
                           (s_part[2][tid] + s_part[3][tid]);
}

extern "C" void kernel_launch(void* const* d_in, const int* in_sizes, int n_in,
                              void* d_out, int out_size, void* d_ws, size_t ws_size,
                              hipStream_t stream) {
  const int*   adj  = (const int*)d_in[0];     // (1,1024,1024) int32
  const float* X    = (const float*)d_in[1];   // (1,1024,32)   f32
  const float* edge = (const float*)d_in[2];   // (1,1024,1024,8) f32
  const float* W    = (const float*)d_in[3];   // (72,64)       f32
  const float* b    = (const float*)d_in[4];   // (64,)         f32
  float* out = (float*)d_out;                  // (1,1024,64)   f32

  float* selfT = (float*)d_ws;                 // 1024*64 floats
  float* nbrT  = selfT + N_NODES * H_OUT;      // 1024*64 floats

  precompute_terms<<<(N_NODES * H_OUT + 255) / 256, 256, 0, stream>>>(
      X, W, b, selfT, nbrT);
  edgeconv_wmma<<<N_NODES, 128, 0, stream>>>(adj, edge, W, selfT, nbrT, out);
}